// GCNConv_59854664237624
// MI455X (gfx1250) — compile-verified
//
#include <hip/hip_runtime.h>

typedef float v2f __attribute__((ext_vector_type(2)));
typedef float v8f __attribute__((ext_vector_type(8)));

#define EMB 128

// ---------------------------------------------------------------------------
// Kernel 1: Y = X @ W   (dense f32 GEMM via V_WMMA_F32_16X16X4_F32)
// Block = 256 threads = 8 waves. Block b handles rows [16b, 16b+16);
// wave w handles output columns [16w, 16w+16). K loop: 32 x (16x16x4) WMMA.
// ---------------------------------------------------------------------------
__global__ __launch_bounds__(256) void gcn_gemm_xw(const float* __restrict__ X,
                                                   const float* __restrict__ W,
                                                   float* __restrict__ Y,
                                                   int n_rows) {
    const int wave = threadIdx.x >> 5;   // 0..7 -> 16-wide column tile
    const int lane = threadIdx.x & 31;
    const int l15  = lane & 15;
    const int hi   = lane >> 4;          // 0 for lanes 0-15, 1 for 16-31
    const int m0   = blockIdx.x * 16;
    const int n0   = wave * 16;

    // A-matrix (16x4 f32): lanes 0-15 hold M=lane with K={k,k+1},
    //                      lanes 16-31 hold M=lane-16 with K={k+2,k+3}.
    int row  = m0 + l15;
    int rowc = row < n_rows ? row : (n_rows - 1);   // clamp loads; stores masked
    const float* Arow = X + (size_t)rowc * EMB;
    const int khalf = hi << 1;                      // 0 or 2

    // B-matrix (4x16 f32): lanes 0-15 hold N=lane with K rows {k,k+1},
    //                      lanes 16-31 hold N=lane-16 with K rows {k+2,k+3}.
    const int ncol = n0 + l15;

    v8f c = {};
#pragma unroll
    for (int k = 0; k < EMB; k += 4) {
        const int kb = k + khalf;                   // even -> 8B aligned
        v2f a = *(const v2f*)(Arow + kb);
        v2f b;
        b.x = W[(size_t)kb * EMB + ncol];
        b.y = W[(size_t)(kb + 1) * EMB + ncol];
        // 8 args: (neg_a, A, neg_b, B, c_mod, C, reuse_a, reuse_b)
        c = __builtin_amdgcn_wmma_f32_16x16x4_f32(false, a, false, b,
                                                  (short)0, c, false, false);
    }

    // C/D layout: VGPR i -> (M = i + 8*hi, N = l15) within the tile.
    const int mBase = m0 + (hi << 3);
#pragma unroll
    for (int i = 0; i < 8; ++i) {
        int m = mBase + i;
        if (m < n_rows) Y[(size_t)m * EMB + n0 + l15] = c[i];
    }
}

// ---------------------------------------------------------------------------
// Kernel 2: zero the output (atomic accumulation target)
// ---------------------------------------------------------------------------
__global__ void gcn_zero(float4* __restrict__ p, long n4) {
    long i = (long)blockIdx.x * blockDim.x + threadIdx.x;
    if (i < n4) p[i] = make_float4(0.f, 0.f, 0.f, 0.f);
}

// ---------------------------------------------------------------------------
// Kernel 3: out[row] += edge_val * Y[col]   (one wave per edge, 4 f32/lane)
// Y gather is L2-resident (51 MB << 192 MB); atomics resolve at L2.
// ---------------------------------------------------------------------------
__global__ __launch_bounds__(256) void gcn_scatter(const float* __restrict__ Y,
                                                   const float* __restrict__ ev,
                                                   const int* __restrict__ er,
                                                   const int* __restrict__ ec,
                                                   float* __restrict__ out,
                                                   int n_edges) {
    const int e = blockIdx.x * (blockDim.x >> 5) + (threadIdx.x >> 5);
    if (e >= n_edges) return;
    const int lane = threadIdx.x & 31;

    const float v = ev[e];
    const int   r = er[e];
    const int   c = ec[e];

    float4 m = ((const float4*)(Y + (size_t)c * EMB))[lane];
    float* o = out + (size_t)r * EMB + lane * 4;
#if defined(__HIP_DEVICE_COMPILE__)
    unsafeAtomicAdd(o + 0, v * m.x);
    unsafeAtomicAdd(o + 1, v * m.y);
    unsafeAtomicAdd(o + 2, v * m.z);
    unsafeAtomicAdd(o + 3, v * m.w);
#endif
}

// ---------------------------------------------------------------------------
// Launch: Y = X@W (ws) ; out = 0 ; out += G @ Y
// ---------------------------------------------------------------------------
extern "C" void kernel_launch(void* const* d_in, const int* in_sizes, int n_in,
                              void* d_out, int out_size, void* d_ws, size_t ws_size,
                              hipStream_t stream) {
    const float* X  = (const float*)d_in[0];
    const float* W  = (const float*)d_in[1];
    const float* ev = (const float*)d_in[2];
    const int*   er = (const int*)d_in[3];
    const int*   ec = (const int*)d_in[4];

    const int n_rows  = in_sizes[0] / EMB;   // 100000
    const int n_edges = in_sizes[2];         // 3200000

    float* out = (float*)d_out;
    float* Y   = (float*)d_ws;               // n_rows * 128 floats (51.2 MB)

    // 1) dense projection first: Y = X @ W  (WMMA f32)
    const int rowTiles = (n_rows + 15) / 16;
    gcn_gemm_xw<<<rowTiles, 256, 0, stream>>>(X, W, Y, n_rows);

    // 2) zero accumulation target
    const long n4 = (long)out_size / 4;
    gcn_zero<<<(int)((n4 + 255) / 256), 256, 0, stream>>>((float4*)out, n4);

    // 3) sparse aggregation into final output: out = G @ Y
    const int wavesPerBlock = 256 / 32;
    const int blocks = (n_edges + wavesPerBlock - 1) / wavesPerBlock;
    gcn_scatter<<<blocks, 256, 0, stream>>>(Y, ev, er, ec, out, n_edges);
}